// BatchDentateCircuit_45741401702622
// MI455X (gfx1250) — compile-verified
//
#include <hip/hip_runtime.h>

#define NB     32
#define NPRE   1000
#define NPOST  1000
#define NP4    250                 // NPOST / 4
#define ICHUNKS 20
#define CHUNK  (NPRE / ICHUNKS)    // 50
#define SPITCH 1008                // padded row stride for S scratch

// exp(-0.1/2) and exp(-0.1/100), and the blended drive coefficients
#define ADEC 0.9512294245007140f
#define NDEC 0.9990004998333750f
#define CA   0.0341394028495002f   // (1-ADEC)*(1-0.3)
#define CN   0.0002998500499875f   // (1-NDEC)*0.3

typedef float f4  __attribute__((ext_vector_type(4)));
typedef float v2f __attribute__((ext_vector_type(2)));
typedef float v8f __attribute__((ext_vector_type(8)));

// ---------------------------------------------------------------------------
// Kernel 1: HBM-bound streaming pass. Each thread: fixed (b, 4 consecutive j),
// loops over an i-chunk. NT hints on the single-use state streams keep the
// reusable g_max/conn (8 MB) resident in L2 across the 32 batches.
// Also accumulates per-thread partial sums of the OLD states for the totals.
// ---------------------------------------------------------------------------
__global__ void __launch_bounds__(256)
syn_stream(const float* __restrict__ pre, const float* __restrict__ mod,
           const float* __restrict__ gmax, const float* __restrict__ conn,
           const float* __restrict__ ampa, const float* __restrict__ nmda,
           float* __restrict__ ampa_out, float* __restrict__ nmda_out,
           float* __restrict__ sa_part, float* __restrict__ sn_part)
{
    const int t = threadIdx.x;
    if (t >= NP4) return;
    const int b  = blockIdx.y;
    const int c  = blockIdx.x;
    const int i0 = c * CHUNK;

    const float modb = mod[b];

    const f4* g4 = (const f4*)gmax + (size_t)i0 * NP4 + t;
    const f4* c4 = (const f4*)conn + (size_t)i0 * NP4 + t;
    const size_t sbase = ((size_t)b * NPRE + i0) * NP4 + t;
    const f4* a4  = (const f4*)ampa + sbase;
    const f4* n4  = (const f4*)nmda + sbase;
    f4* ao4 = (f4*)ampa_out + sbase;
    f4* no4 = (f4*)nmda_out + sbase;
    const float* pr = pre + (size_t)b * NPRE + i0;

    f4 accA = {0.f, 0.f, 0.f, 0.f};
    f4 accN = {0.f, 0.f, 0.f, 0.f};

#pragma unroll 2
    for (int i = 0; i < CHUNK; ++i) {
        const size_t off = (size_t)i * NP4;
        f4 g  = g4[off];
        f4 cc = c4[off];
        f4 av = __builtin_nontemporal_load(&a4[off]);
        f4 nv = __builtin_nontemporal_load(&n4[off]);

        const float pm   = pr[i] * modb;      // uniform (SALU-friendly)
        const float capm = CA * pm;
        const float cnpm = CN * pm;

        f4 w = g * cc;                        // effective weight
        f4 ao, no;
        ao.x = __builtin_fmaf(ADEC, av.x, capm * w.x);
        ao.y = __builtin_fmaf(ADEC, av.y, capm * w.y);
        ao.z = __builtin_fmaf(ADEC, av.z, capm * w.z);
        ao.w = __builtin_fmaf(ADEC, av.w, capm * w.w);
        no.x = __builtin_fmaf(NDEC, nv.x, cnpm * w.x);
        no.y = __builtin_fmaf(NDEC, nv.y, cnpm * w.y);
        no.z = __builtin_fmaf(NDEC, nv.z, cnpm * w.z);
        no.w = __builtin_fmaf(NDEC, nv.w, cnpm * w.w);

        accA += av;                           // sums of OLD state
        accN += nv;

        __builtin_nontemporal_store(ao, &ao4[off]);
        __builtin_nontemporal_store(no, &no4[off]);
    }

    const size_t pidx = ((size_t)c * NB + b) * NP4 + t;
    ((f4*)sa_part)[pidx] = accA;
    ((f4*)sn_part)[pidx] = accN;
}

// ---------------------------------------------------------------------------
// Kernel 2: S = (pre .* mod) @ (g_max .* conn)   -- M=32, N=1000, K=1000, f32
// One wave per 16x16 tile, V_WMMA_F32_16X16X4_F32 over 250 K-steps.
// Operand layouts per CDNA5 ISA 7.12.2:
//   A (16x4):  lanes 0-15 -> K={k0,k0+1}, lanes 16-31 -> K={k0+2,k0+3}, M=lane%16
//   B (4x16):  mirrored striping, N=lane%16
//   C/D:       8 VGPRs, M = r + 8*(lane/16), N = lane%16
// EXEC stays all-ones; out-of-range columns are value-masked to 0.
// ---------------------------------------------------------------------------
__global__ void __launch_bounds__(32)
syn_gemm_wmma(const float* __restrict__ pre, const float* __restrict__ mod,
              const float* __restrict__ gmax, const float* __restrict__ conn,
              float* __restrict__ S)
{
    const int lane = threadIdx.x;          // 0..31
    const int nt   = blockIdx.x;           // 0..62 (N tiles of 16)
    const int mt   = blockIdx.y;           // 0..1  (M tiles of 16)

    const int row   = mt * 16 + (lane & 15);            // batch row for A
    const int jcol  = nt * 16 + (lane & 15);            // column for B
    const int jc    = (jcol < NPOST) ? jcol : (NPOST - 1);
    const float jmask = (jcol < NPOST) ? 1.0f : 0.0f;
    const int khalf = (lane >> 4) * 2;                  // 0 or 2

    const float modr = mod[row];
    const float* prow = pre + (size_t)row * NPRE;

    v8f acc = {0.f, 0.f, 0.f, 0.f, 0.f, 0.f, 0.f, 0.f};

    for (int k0 = 0; k0 < NPRE; k0 += 4) {
        const int ka = k0 + khalf;
        v2f A;
        A[0] = prow[ka]     * modr;
        A[1] = prow[ka + 1] * modr;
        v2f Bv;
        Bv[0] = gmax[(size_t)ka * NPOST + jc]       * conn[(size_t)ka * NPOST + jc]       * jmask;
        Bv[1] = gmax[(size_t)(ka + 1) * NPOST + jc] * conn[(size_t)(ka + 1) * NPOST + jc] * jmask;

        // (neg_a, A, neg_b, B, c_mod, C, reuse_a, reuse_b)
        acc = __builtin_amdgcn_wmma_f32_16x16x4_f32(
            false, A, false, Bv, (short)0, acc, false, false);
    }

    const int ncol  = nt * 16 + (lane & 15);
    const int mbase = mt * 16 + (lane >> 4) * 8;
#pragma unroll
    for (int r = 0; r < 8; ++r)
        S[(size_t)(mbase + r) * SPITCH + ncol] = acc[r];
}

// ---------------------------------------------------------------------------
// Kernel 3: totals[b,j] = decay * sum_i(old_state) + coef * S[b,j]
// ---------------------------------------------------------------------------
__global__ void __launch_bounds__(256)
syn_totals(const float* __restrict__ sa_part, const float* __restrict__ sn_part,
           const float* __restrict__ S,
           float* __restrict__ ta, float* __restrict__ tn)
{
    const int idx = blockIdx.x * blockDim.x + threadIdx.x;   // over B*NPOST
    if (idx >= NB * NPOST) return;
    const int b = idx / NPOST;
    const int j = idx - b * NPOST;

    float sa = 0.f, sn = 0.f;
#pragma unroll
    for (int c = 0; c < ICHUNKS; ++c) {
        sa += sa_part[(size_t)c * NB * NPOST + idx];
        sn += sn_part[(size_t)c * NB * NPOST + idx];
    }
    const float s = S[(size_t)b * SPITCH + j];
    ta[idx] = __builtin_fmaf(ADEC, sa, CA * s);
    tn[idx] = __builtin_fmaf(NDEC, sn, CN * s);
}

// ---------------------------------------------------------------------------
extern "C" void kernel_launch(void* const* d_in, const int* in_sizes, int n_in,
                              void* d_out, int out_size, void* d_ws, size_t ws_size,
                              hipStream_t stream)
{
    const float* pre  = (const float*)d_in[0];   // [B, NPRE]
    const float* mod  = (const float*)d_in[1];   // [B]
    const float* gmax = (const float*)d_in[2];   // [NPRE, NPOST]
    const float* conn = (const float*)d_in[3];   // [NPRE, NPOST]
    const float* ampa = (const float*)d_in[4];   // [B, NPRE, NPOST]
    const float* nmda = (const float*)d_in[5];   // [B, NPRE, NPOST]

    float* out = (float*)d_out;
    const size_t nstate = (size_t)NB * NPRE * NPOST;     // 32,000,000
    float* ampa_out = out;
    float* nmda_out = out + nstate;
    float* ta       = out + 2 * nstate;
    float* tn       = ta + (size_t)NB * NPOST;

    float* ws      = (float*)d_ws;
    const size_t npart = (size_t)ICHUNKS * NB * NPOST;   // 640,000
    float* sa_part = ws;
    float* sn_part = ws + npart;
    float* Sbuf    = ws + 2 * npart;                     // [32, SPITCH]

    dim3 g1(ICHUNKS, NB);
    syn_stream<<<g1, 256, 0, stream>>>(pre, mod, gmax, conn, ampa, nmda,
                                       ampa_out, nmda_out, sa_part, sn_part);

    dim3 g2((NPOST + 15) / 16, NB / 16);                 // 63 x 2 tiles
    syn_gemm_wmma<<<g2, 32, 0, stream>>>(pre, mod, gmax, conn, Sbuf);

    syn_totals<<<(NB * NPOST + 255) / 256, 256, 0, stream>>>(sa_part, sn_part,
                                                             Sbuf, ta, tn);
}